// SamplePatches_2156073583006
// MI455X (gfx1250) — compile-verified
//
#include <hip/hip_runtime.h>
#include <hip/hip_bf16.h>
#include <stdint.h>

// ---------------------------------------------------------------------------
// Problem constants (from reference)
// ---------------------------------------------------------------------------
#define BATCH      8
#define HS         192
#define WS         192
#define ATT_N      (HS * WS)          // 36864 per batch
#define TOTAL      (BATCH * ATT_N)    // 294912
#define HALF       (TOTAL / 2)        // 147456
#define NPATCH     32
#define PATCH      100
#define HHIGH      1792
#define WHIGH      1792
#define CH         3
#define PATCH_ELEMS (PATCH * PATCH)   // 10000
#define OUT_PATCH_FLOATS (BATCH * NPATCH * CH * PATCH_ELEMS) // 7,680,000

typedef unsigned int u32;
typedef unsigned long long u64;
typedef __attribute__((ext_vector_type(4))) unsigned int u32x4;
typedef __attribute__((ext_vector_type(8))) unsigned int u32x8;

// ---------------------------------------------------------------------------
// threefry2x32 (JAX default PRNG), key = (0, 42) from jax.random.key(42)
// ---------------------------------------------------------------------------
__device__ __forceinline__ u32 rotl32(u32 x, u32 n) {
    return (x << n) | (x >> (32u - n));
}

__device__ __forceinline__ void threefry2x32(u32 k0, u32 k1, u32 c0, u32 c1,
                                             u32& o0, u32& o1) {
    const u32 ks0 = k0, ks1 = k1, ks2 = k0 ^ k1 ^ 0x1BD11BDAu;
    u32 x0 = c0 + ks0;
    u32 x1 = c1 + ks1;
#define TF_ROUND(r) { x0 += x1; x1 = rotl32(x1, (r)); x1 ^= x0; }
    TF_ROUND(13) TF_ROUND(15) TF_ROUND(26) TF_ROUND(6)
    x0 += ks1; x1 += ks2 + 1u;
    TF_ROUND(17) TF_ROUND(29) TF_ROUND(16) TF_ROUND(24)
    x0 += ks2; x1 += ks0 + 2u;
    TF_ROUND(13) TF_ROUND(15) TF_ROUND(26) TF_ROUND(6)
    x0 += ks0; x1 += ks1 + 3u;
    TF_ROUND(17) TF_ROUND(29) TF_ROUND(16) TF_ROUND(24)
    x0 += ks1; x1 += ks2 + 4u;
    TF_ROUND(13) TF_ROUND(15) TF_ROUND(26) TF_ROUND(6)
    x0 += ks2; x1 += ks0 + 5u;
#undef TF_ROUND
    o0 = x0;
    o1 = x1;
}

// bits -> uniform[tiny,1) -> gumbel, matching jax.random.gumbel
__device__ __forceinline__ float bits_to_gumbel(u32 bits) {
    const float tiny = 1.17549435082228750797e-38f; // float32 finfo.tiny
    u32 fb = (bits >> 9) | 0x3F800000u;
    float f = __uint_as_float(fb) - 1.0f;           // [0,1)
    float u = f * (1.0f - tiny) + tiny;
    u = fmaxf(tiny, u);
    return -logf(-logf(u));
}

// ---------------------------------------------------------------------------
// Kernel 1: scores[pos] = log(att[pos]) + gumbel(bits[pos])
// JAX counter layout: bits[i] = out0(i), bits[i+HALF] = out1(i), counters
// (i, i+HALF), i in [0, HALF).
// ---------------------------------------------------------------------------
__global__ void sp_gumbel_scores(const float* __restrict__ att,
                                 float* __restrict__ scores) {
    u32 i = blockIdx.x * blockDim.x + threadIdx.x;
    if (i >= HALF) return;
    u32 o0, o1;
    threefry2x32(0u, 42u, i, i + (u32)HALF, o0, o1);
    scores[i]        = logf(att[i])        + bits_to_gumbel(o0);
    scores[i + HALF] = logf(att[i + HALF]) + bits_to_gumbel(o1);
}

// ---------------------------------------------------------------------------
// Kernel 2: per-batch iterative argmax top-32 (ties -> lowest index),
// records start coordinates and sampled attention.
// start = (16 + coord) * 8 - 50, clipped to [0, 1792-100]  (exact: see analysis)
// ---------------------------------------------------------------------------
__global__ void sp_topk(const float* __restrict__ att,
                        float* __restrict__ scores,   // mutated (masked), rebuilt by k1 each call
                        int* __restrict__ start_y,
                        int* __restrict__ start_x,
                        float* __restrict__ samp_out) {
    __shared__ float redv[256];
    __shared__ int   redi[256];
    const int b = blockIdx.x;
    const int t = threadIdx.x;
    float* sc = scores + (size_t)b * ATT_N;

    for (int k = 0; k < NPATCH; ++k) {
        float best = -__builtin_inff();
        int   bi   = 0x7FFFFFFF;
        for (int j = t; j < ATT_N; j += 256) {
            float v = sc[j];
            if (v > best) { best = v; bi = j; }
        }
        redv[t] = best;
        redi[t] = bi;
        __syncthreads();
        for (int s = 128; s > 0; s >>= 1) {
            if (t < s) {
                float v2 = redv[t + s]; int i2 = redi[t + s];
                if (v2 > redv[t] || (v2 == redv[t] && i2 < redi[t])) {
                    redv[t] = v2; redi[t] = i2;
                }
            }
            __syncthreads();
        }
        if (t == 0) {
            int idx = redi[0];
            sc[idx] = -__builtin_inff();          // mask for next pass
            int y = idx / WS;
            int x = idx % WS;
            int syv = (16 + y) * 8 - 50;
            int sxv = (16 + x) * 8 - 50;
            syv = min(max(syv, 0), HHIGH - PATCH);
            sxv = min(max(sxv, 0), WHIGH - PATCH);
            start_y[b * NPATCH + k] = syv;
            start_x[b * NPATCH + k] = sxv;
            samp_out[b * NPATCH + k] = att[(size_t)b * ATT_N + idx];
        }
        __syncthreads();
    }
}

// ---------------------------------------------------------------------------
// Kernel 3: one wave per (b, patch, channel) tile. Build TDM descriptors in
// SGPRs and DMA the 100x100 f32 tile: global (stride 1792) -> LDS -> global
// (contiguous). CDNA5 Tensor Data Mover path (CDNA5 ISA).
// ---------------------------------------------------------------------------
__device__ __forceinline__ void sp_wait_tensorcnt0() {
#if defined(__has_builtin)
#if __has_builtin(__builtin_amdgcn_s_wait_tensorcnt)
    __builtin_amdgcn_s_wait_tensorcnt(0);
    return;
#endif
#endif
    asm volatile("s_wait_tensorcnt 0x0" ::: "memory");
}

__global__ __launch_bounds__(32) void sp_patch_tdm(const float* __restrict__ x_high,
                                                   const int* __restrict__ start_y,
                                                   const int* __restrict__ start_x,
                                                   float* __restrict__ out) {
    __shared__ float tile[PATCH_ELEMS];

    const int blk = blockIdx.x;          // ((b*32 + k)*3 + c)
    const int b   = blk / (NPATCH * CH);
    const int rem = blk % (NPATCH * CH);
    const int k   = rem / CH;
    const int c   = rem % CH;

    const int syv = __builtin_amdgcn_readfirstlane(start_y[b * NPATCH + k]);
    const int sxv = __builtin_amdgcn_readfirstlane(start_x[b * NPATCH + k]);

    const u64 src = (u64)(uintptr_t)x_high +
        4ull * ((((u64)b * CH + (u64)c) * (u64)HHIGH + (u64)syv) * (u64)WHIGH + (u64)sxv);
    const u64 dst = (u64)(uintptr_t)out + 4ull * (u64)PATCH_ELEMS * (u64)blk;

    // LDS byte offset of the tile (generic LDS pointer: low 32 bits = offset)
    u32 lds_off = (u32)(uintptr_t)(void*)&tile[0];
    lds_off = __builtin_amdgcn_readfirstlane(lds_off);

    // ---- D# group 0: count=1, lds_addr, global_addr[56:0], type=2 ----
    u32x4 g0_load, g0_store;
    g0_load.x = 1u;                                   // count=1, user mode
    g0_load.y = lds_off;                              // lds_addr
    g0_load.z = (u32)(src & 0xFFFFFFFFull);           // global_addr[31:0]
    g0_load.w = ((u32)(src >> 32) & 0x01FFFFFFu) | (2u << 30); // [56:32] | type=2

    g0_store.x = 1u;
    g0_store.y = lds_off;
    g0_store.z = (u32)(dst & 0xFFFFFFFFull);
    g0_store.w = ((u32)(dst >> 32) & 0x01FFFFFFu) | (2u << 30);

    // ---- D# group 1 (256 bits), data_size=2 (4B elements), 2D tile ----
    // word0: workgroup_mask=0 | data_size<<16 | flags=0
    // tensor_dim0  @ bits [79:48]   tensor_dim1 @ [111:80]
    // tile_dim0    @ bits [127:112] tile_dim1   @ [143:128] tile_dim2 @ [159:144]
    // tensor_dim0_stride @ [207:160] (48b)      tensor_dim1_stride @ [255:208]
    u32x8 g1_load;
    g1_load[0] = (2u << 16);                  // data_size = 4 bytes
    g1_load[1] = ((u32)WHIGH & 0xFFFFu) << 16;          // tensor_dim0 lo16
    g1_load[2] = ((u32)WHIGH >> 16) | (((u32)HHIGH & 0xFFFFu) << 16); // dim0 hi | dim1 lo
    g1_load[3] = ((u32)HHIGH >> 16) | ((u32)PATCH << 16);             // dim1 hi | tile_dim0
    g1_load[4] = (u32)PATCH;                  // tile_dim1 (tile_dim2 = 0)
    g1_load[5] = (u32)WHIGH;                  // tensor_dim0_stride[31:0] = 1792
    g1_load[6] = 0u;                          // stride hi16 | dim1_stride lo16
    g1_load[7] = 0u;

    u32x8 g1_store;
    g1_store[0] = (2u << 16);
    g1_store[1] = ((u32)PATCH & 0xFFFFu) << 16;         // tensor_dim0 = 100
    g1_store[2] = ((u32)PATCH & 0xFFFFu) << 16;         // tensor_dim1 = 100
    g1_store[3] = ((u32)PATCH << 16);                   // tile_dim0 = 100
    g1_store[4] = (u32)PATCH;                           // tile_dim1 = 100
    g1_store[5] = (u32)PATCH;                           // dim0 stride = 100 (contiguous)
    g1_store[6] = 0u;
    g1_store[7] = 0u;

    const u32x4 gz = {0u, 0u, 0u, 0u};        // groups 2/3: unused dims zero

    // DMA: global -> LDS
    asm volatile("tensor_load_to_lds %0, %1, %2, %3"
                 :
                 : "s"(g0_load), "s"(g1_load), "s"(gz), "s"(gz)
                 : "memory");
    sp_wait_tensorcnt0();                      // tile resident in LDS

    // DMA: LDS -> global (contiguous patch in output)
    asm volatile("tensor_store_from_lds %0, %1, %2, %3"
                 :
                 : "s"(g0_store), "s"(g1_store), "s"(gz), "s"(gz)
                 : "memory");
    sp_wait_tensorcnt0();                      // drain before wave exit
}

// ---------------------------------------------------------------------------
// Host-side launcher
// ---------------------------------------------------------------------------
extern "C" void kernel_launch(void* const* d_in, const int* in_sizes, int n_in,
                              void* d_out, int out_size, void* d_ws, size_t ws_size,
                              hipStream_t stream) {
    // inputs: x_low (unused beyond shape), x_high, attention
    const float* x_high    = (const float*)d_in[1];
    const float* attention = (const float*)d_in[2];
    float* out = (float*)d_out;

    // workspace layout
    float* scores  = (float*)d_ws;                                  // 294912 f32
    int*   start_y = (int*)((char*)d_ws + (size_t)TOTAL * sizeof(float));
    int*   start_x = start_y + BATCH * NPATCH;

    // 1) deterministic gumbel + log(att) scores (threefry key (0,42))
    sp_gumbel_scores<<<(HALF + 255) / 256, 256, 0, stream>>>(attention, scores);

    // 2) per-batch top-32 (descending), sampled attention to output tail
    sp_topk<<<BATCH, 256, 0, stream>>>(attention, scores, start_y, start_x,
                                       out + OUT_PATCH_FLOATS);

    // 3) TDM tile copies: one wave per (b, patch, channel)
    sp_patch_tdm<<<BATCH * NPATCH * CH, 32, 0, stream>>>(x_high, start_y, start_x, out);
}